// EMM_NTM_51453708206498
// MI455X (gfx1250) — compile-verified
//
#include <hip/hip_runtime.h>
#include <hip/hip_bf16.h>

#define Bn 64
#define Hn 1024
#define Nn 65536
#define Mn 64
#define Sn 3
#define Rn 4

typedef float v2f __attribute__((ext_vector_type(2)));
typedef float v8f __attribute__((ext_vector_type(8)));

static __device__ __forceinline__ v8f wmma_f32(v2f a, v2f b, v8f c) {
  // D = A(16x4) * B(4x16) + C(16x16), fp32, wave32
  return __builtin_amdgcn_wmma_f32_16x16x4_f32(false, a, false, b, (short)0, c,
                                               false, false);
}

// ---------------------------------------------------------------------------
// Head projections: k, kn, beta, g, gamma, s(softmax), e, a  (per batch row)
// ---------------------------------------------------------------------------
__global__ void head_params_kernel(
    const float* __restrict__ h, const float* __restrict__ kw,
    const float* __restrict__ kb, const float* __restrict__ bw,
    const float* __restrict__ bb, const float* __restrict__ gw,
    const float* __restrict__ gb, const float* __restrict__ sw,
    const float* __restrict__ sb, const float* __restrict__ gmw,
    const float* __restrict__ gmb, const float* __restrict__ ew,
    const float* __restrict__ ebias, const float* __restrict__ aw,
    const float* __restrict__ abias, float* __restrict__ kn,
    float* __restrict__ beta, float* __restrict__ g, float* __restrict__ gamma,
    float* __restrict__ s, float* __restrict__ e_out, float* __restrict__ a_out) {
  __shared__ float sh[Hn];
  __shared__ float raw[224];
  __shared__ float kbuf[Mn];
  __shared__ float nrm;
  const int b = blockIdx.x;
  const int t = threadIdx.x;
  for (int i = t; i < Hn; i += 256) sh[i] = h[b * Hn + i];
  __syncthreads();
  if (t < 198) {
    float acc = 0.f;
    if (t < 64) {
      for (int i = 0; i < Hn; ++i) acc += sh[i] * kw[i * Mn + t];
    } else if (t < 128) {
      const int m = t - 64;
      for (int i = 0; i < Hn; ++i) acc += sh[i] * ew[i * Mn + m];
    } else if (t < 192) {
      const int m = t - 128;
      for (int i = 0; i < Hn; ++i) acc += sh[i] * aw[i * Mn + m];
    } else if (t < 195) {
      const int j = t - 192;
      for (int i = 0; i < Hn; ++i) acc += sh[i] * sw[i * Sn + j];
    } else if (t == 195) {
      for (int i = 0; i < Hn; ++i) acc += sh[i] * bw[i];
    } else if (t == 196) {
      for (int i = 0; i < Hn; ++i) acc += sh[i] * gw[i];
    } else {
      for (int i = 0; i < Hn; ++i) acc += sh[i] * gmw[i];
    }
    raw[t] = acc;
  }
  __syncthreads();
  if (t < 64) {
    float k = raw[t] + kb[t];
    kbuf[t] = fminf(fmaxf(k, 0.f), 1.f);
  }
  __syncthreads();
  if (t == 0) {
    float ss = 0.f;
    for (int i = 0; i < Mn; ++i) ss += kbuf[i] * kbuf[i];
    nrm = sqrtf(ss) + 1e-8f;
    beta[b] = fmaxf(raw[195] + bb[0], 0.f);
    g[b] = fminf(fmaxf(raw[196] + gb[0], 0.f), 1.f);
    gamma[b] = 1.f + fmaxf(raw[197] + gmb[0], 0.f);
    float s0 = raw[192] + sb[0], s1 = raw[193] + sb[1], s2 = raw[194] + sb[2];
    float mx = fmaxf(s0, fmaxf(s1, s2));
    float e0 = __expf(s0 - mx), e1 = __expf(s1 - mx), e2 = __expf(s2 - mx);
    float inv = 1.f / (e0 + e1 + e2);
    s[b * 3 + 0] = e0 * inv;
    s[b * 3 + 1] = e1 * inv;
    s[b * 3 + 2] = e2 * inv;
  }
  __syncthreads();
  if (t < 64) {
    kn[b * Mn + t] = kbuf[t] / nrm;
  } else if (t < 128) {
    const int m = t - 64;
    float v = raw[t] + ebias[m];
    e_out[b * Mn + m] = fminf(fmaxf(v, 0.f), 1.f);
  } else if (t < 192) {
    const int m = t - 128;
    float v = raw[t] + abias[m];
    a_out[b * Mn + m] = fminf(fmaxf(v, 0.f), 1.f);
  }
}

// ---------------------------------------------------------------------------
// Row inverse norms of memory: invnorm[n] = 1/(||m[n,:]|| + eps)
// ---------------------------------------------------------------------------
__global__ void rownorm_kernel(const float* __restrict__ mem,
                               const int* __restrict__ bank_no,
                               float* __restrict__ invnorm) {
  const int bank = bank_no ? bank_no[0] : 0;
  const float* m = mem + (size_t)bank * Nn * Mn;
  const int t = threadIdx.x;
  const int rowl = t >> 2, part = t & 3;
  const int n = blockIdx.x * 64 + rowl;
  const float4* p = (const float4*)(m + (size_t)n * Mn) + part * 4;
  float ss = 0.f;
#pragma unroll
  for (int i = 0; i < 4; ++i) {
    float4 v = p[i];
    ss += v.x * v.x + v.y * v.y + v.z * v.z + v.w * v.w;
  }
  ss += __shfl_xor(ss, 1);
  ss += __shfl_xor(ss, 2);
  if (part == 0) invnorm[n] = 1.f / (sqrtf(ss) + 1e-8f);
}

// ---------------------------------------------------------------------------
// z[b,n] = beta[b] * (kn[b,:] . m[n,:]*invnorm[n])  via V_WMMA_F32_16X16X4_F32
// block: 256 thr (8 waves); wave handles 16 columns, all 64 batch rows
// ---------------------------------------------------------------------------
__global__ void z_wmma_kernel(const float* __restrict__ mem,
                              const int* __restrict__ bank_no,
                              const float* __restrict__ kn,
                              const float* __restrict__ invnorm,
                              const float* __restrict__ beta,
                              float* __restrict__ z) {
  __shared__ float skn[Bn * Mn];  // 16 KB
  __shared__ float sbeta[Bn];
  const int bank = bank_no ? bank_no[0] : 0;
  const float* m = mem + (size_t)bank * Nn * Mn;
  const int t = threadIdx.x;
  for (int i = t; i < Bn * Mn; i += 256) skn[i] = kn[i];
  if (t < Bn) sbeta[t] = beta[t];
  __syncthreads();
  const int wave = t >> 5, lane = t & 31;
  const int n0 = blockIdx.x * 128 + wave * 16;
  const int col = n0 + (lane & 15);
  const int khalf = (lane >> 4) * 2;  // lanes 0-15 -> K0/K1, lanes 16-31 -> K2/K3
  const float inv = invnorm[col];
  const float* mrow = m + (size_t)col * Mn;
  v8f c0 = {}, c1 = {}, c2 = {}, c3 = {};
  const int r = lane & 15;
#pragma unroll
  for (int kt = 0; kt < 16; ++kt) {
    const int kk = kt * 4 + khalf;
    v2f bf;
    bf.x = mrow[kk] * inv;
    bf.y = mrow[kk + 1] * inv;
    v2f a0, a1, a2, a3;
    a0.x = skn[r * Mn + kk];        a0.y = skn[r * Mn + kk + 1];
    a1.x = skn[(r + 16) * Mn + kk]; a1.y = skn[(r + 16) * Mn + kk + 1];
    a2.x = skn[(r + 32) * Mn + kk]; a2.y = skn[(r + 32) * Mn + kk + 1];
    a3.x = skn[(r + 48) * Mn + kk]; a3.y = skn[(r + 48) * Mn + kk + 1];
    c0 = wmma_f32(a0, bf, c0);
    c1 = wmma_f32(a1, bf, c1);
    c2 = wmma_f32(a2, bf, c2);
    c3 = wmma_f32(a3, bf, c3);
  }
  const int half = lane >> 4;
  const int nc = n0 + (lane & 15);
#pragma unroll
  for (int j = 0; j < 8; ++j) {
    const int b0 = j + half * 8;  // C layout: VGPR j -> rows j (lanes 0-15), j+8 (16-31)
    z[(size_t)(b0)      * Nn + nc] = c0[j] * sbeta[b0];
    z[(size_t)(b0 + 16) * Nn + nc] = c1[j] * sbeta[b0 + 16];
    z[(size_t)(b0 + 32) * Nn + nc] = c2[j] * sbeta[b0 + 32];
    z[(size_t)(b0 + 48) * Nn + nc] = c3[j] * sbeta[b0 + 48];
  }
}

// ---------------------------------------------------------------------------
// Softmax reductions (per-row over N, chunked 16 blocks/row)
// ---------------------------------------------------------------------------
__global__ void rowmax_partial_kernel(const float* __restrict__ z,
                                      float* __restrict__ part) {
  __shared__ float red[256];
  const int b = blockIdx.y, cx = blockIdx.x, t = threadIdx.x;
  const float* row = z + (size_t)b * Nn + cx * 4096;
  float mx = -3.4e38f;
#pragma unroll
  for (int i = 0; i < 16; ++i) mx = fmaxf(mx, row[t + i * 256]);
  red[t] = mx;
  __syncthreads();
  for (int s2 = 128; s2 > 0; s2 >>= 1) {
    if (t < s2) red[t] = fmaxf(red[t], red[t + s2]);
    __syncthreads();
  }
  if (t == 0) part[b * 16 + cx] = red[0];
}

__global__ void row_finalize_kernel(const float* __restrict__ part,
                                    float* __restrict__ out, int op) {
  const int b = threadIdx.x;  // 64 threads
  float v = op ? 0.f : -3.4e38f;
  for (int i = 0; i < 16; ++i) {
    float p = part[b * 16 + i];
    v = op ? (v + p) : fmaxf(v, p);
  }
  out[b] = v;
}

__global__ void expsum_kernel(float* __restrict__ z,
                              const float* __restrict__ rowmax,
                              float* __restrict__ part) {
  __shared__ float red[256];
  const int b = blockIdx.y, cx = blockIdx.x, t = threadIdx.x;
  const float mx = rowmax[b];
  float* row = z + (size_t)b * Nn + cx * 4096;
  float s = 0.f;
#pragma unroll
  for (int i = 0; i < 16; ++i) {
    float p = __expf(row[t + i * 256] - mx);
    row[t + i * 256] = p;
    s += p;
  }
  red[t] = s;
  __syncthreads();
  for (int s2 = 128; s2 > 0; s2 >>= 1) {
    if (t < s2) red[t] += red[t + s2];
    __syncthreads();
  }
  if (t == 0) part[b * 16 + cx] = red[0];
}

// ---------------------------------------------------------------------------
// Fused: gate interpolation + S=3 shift + power sharpening, + global-sum parts
// ---------------------------------------------------------------------------
__global__ void shift_pow_kernel(const float* __restrict__ p,
                                 const float* __restrict__ wprev,
                                 const float* __restrict__ g,
                                 const float* __restrict__ s,
                                 const float* __restrict__ gamma,
                                 const float* __restrict__ rowsum,
                                 float* __restrict__ wpow,
                                 float* __restrict__ gpart) {
  __shared__ float red[256];
  const int b = blockIdx.y, cx = blockIdx.x, t = threadIdx.x;
  const float gb = g[b], omg = 1.f - gb;
  const float s0 = s[b * 3], s1 = s[b * 3 + 1], s2 = s[b * 3 + 2];
  const float gam = gamma[b];
  const float invsum = 1.f / rowsum[b];
  const float* prow = p + (size_t)b * Nn;
  const float* wrow = wprev + (size_t)b * Nn;
  float acc = 0.f;
#pragma unroll 4
  for (int i = 0; i < 16; ++i) {
    const int n = cx * 4096 + t + i * 256;
    float wgm = (n > 0) ? (gb * prow[n - 1] * invsum + omg * wrow[n - 1]) : 0.f;
    float wg0 = gb * prow[n] * invsum + omg * wrow[n];
    float wgp = (n < Nn - 1) ? (gb * prow[n + 1] * invsum + omg * wrow[n + 1]) : 0.f;
    float wt = s0 * wgm + s1 * wg0 + s2 * wgp;
    float wp = powf(wt, gam);
    wpow[(size_t)b * Nn + n] = wp;
    acc += wp;
  }
  red[t] = acc;
  __syncthreads();
  for (int s2_ = 128; s2_ > 0; s2_ >>= 1) {
    if (t < s2_) red[t] += red[t + s2_];
    __syncthreads();
  }
  if (t == 0) gpart[b * 16 + cx] = red[0];
}

__global__ void gsum_kernel(const float* __restrict__ gpart,
                            float* __restrict__ gsum) {
  __shared__ float red[256];
  const int t = threadIdx.x;
  float s = 0.f;
  for (int i = t; i < 1024; i += 256) s += gpart[i];
  red[t] = s;
  __syncthreads();
  for (int s2 = 128; s2 > 0; s2 >>= 1) {
    if (t < s2) red[t] += red[t + s2];
    __syncthreads();
  }
  if (t == 0) gsum[0] = red[0];
}

// ---------------------------------------------------------------------------
// m1 = m0*(1 - ww^T e) + ww^T a   (LDS-tiled, normalization folded in)
// ---------------------------------------------------------------------------
__global__ void mem_update_kernel(const float* __restrict__ mem,
                                  const int* __restrict__ bank_no,
                                  const float* __restrict__ wpow,
                                  const float* __restrict__ gsum,
                                  const float* __restrict__ e,
                                  const float* __restrict__ a,
                                  float* __restrict__ m1) {
  __shared__ float wt[Bn * Mn];
  __shared__ float se[Bn * Mn];
  __shared__ float sa[Bn * Mn];
  const int bank = bank_no ? bank_no[0] : 0;
  const float* m0 = mem + (size_t)bank * Nn * Mn;
  const int t = threadIdx.x;
  const int n0 = blockIdx.x * 64;
  const float invg = 1.f / (gsum[0] + 1e-5f);
  for (int i = t; i < Bn * Mn; i += 256) {
    se[i] = e[i];
    sa[i] = a[i];
  }
  const int nl = t & 63;
  for (int bb = t >> 6; bb < Bn; bb += 4)
    wt[bb * 64 + nl] = wpow[(size_t)bb * Nn + n0 + nl] * invg;
  __syncthreads();
  for (int idx = t; idx < 4096; idx += 256) {
    const int nloc = idx >> 6, c = idx & 63;
    float ae = 0.f, aa = 0.f;
#pragma unroll 8
    for (int bb = 0; bb < Bn; ++bb) {
      const float w = wt[bb * 64 + nloc];
      ae += w * se[bb * 64 + c];
      aa += w * sa[bb * 64 + c];
    }
    const size_t gi = (size_t)(n0 + nloc) * Mn + c;
    m1[gi] = m0[gi] * (1.f - ae) + aa;
  }
}

// ---------------------------------------------------------------------------
// r[b,m] = sum_n w[b,n]/(gsum+eps) * m1[n,m]  — split-K WMMA.
// 64 blocks x 256 thr (8 waves); each block owns a 1024-row K chunk processed
// in 8 stages of 128. Each stage DMAs m1[128x64] + wpow[64x128] tiles into LDS
// via global_load_async_to_lds_b128 (ASYNCcnt), then feeds WMMAs from LDS.
// Wave w -> col tile (w&3), b-tiles {2*(w>>2), 2*(w>>2)+1}. Atomic combine.
// ---------------------------------------------------------------------------
__global__ void read_wmma_kernel(const float* __restrict__ wpow,
                                 const float* __restrict__ gsum,
                                 const float* __restrict__ m1,
                                 float* __restrict__ out, int rhead) {
  __shared__ float sm1[128 * Mn];   // [k][c]  32 KB
  __shared__ float swp[Bn * 128];   // [b][k]  32 KB
  const int t = threadIdx.x;
  const int wave = t >> 5, lane = t & 31;
  const int r = lane & 15;
  const int khalf = (lane >> 4) * 2;
  const int c0 = (wave & 3) * 16;
  const int bt0 = (wave >> 2) * 2;  // this wave's b-tiles: bt0, bt0+1
  const int kchunk0 = blockIdx.x * 1024;
  const uint32_t sm1_base = (uint32_t)(uintptr_t)sm1;  // LDS byte offset
  const uint32_t swp_base = (uint32_t)(uintptr_t)swp;
  v8f acc0 = {}, acc1 = {};
  for (int st = 0; st < 8; ++st) {
    const int k0 = kchunk0 + st * 128;
    // async DMA stage: 2048 float4 per tile, 8 b128 per thread per tile
#pragma unroll
    for (int p = 0; p < 8; ++p) {
      const int li = t + p * 256;  // float4 index in [0,2048)
      const float* ga = m1 + (size_t)k0 * Mn + (size_t)li * 4;
      const uint32_t la = sm1_base + (uint32_t)li * 16u;
      asm volatile("global_load_async_to_lds_b128 %0, %1, off" ::"v"(la),
                   "v"(ga)
                   : "memory");
      const int bb = li >> 5, j = li & 31;  // 32 float4 per wpow row
      const float* gw = wpow + (size_t)bb * Nn + k0 + j * 4;
      const uint32_t lw = swp_base + (uint32_t)(bb * 128 + j * 4) * 4u;
      asm volatile("global_load_async_to_lds_b128 %0, %1, off" ::"v"(lw),
                   "v"(gw)
                   : "memory");
    }
    asm volatile("s_wait_asynccnt 0" ::: "memory");
    __syncthreads();
#pragma unroll
    for (int kt = 0; kt < 32; ++kt) {
      const int kk = kt * 4 + khalf;
      v2f bf, a0, a1;
      bf.x = sm1[kk * Mn + c0 + r];
      bf.y = sm1[(kk + 1) * Mn + c0 + r];
      a0.x = swp[(bt0 * 16 + r) * 128 + kk];
      a0.y = swp[(bt0 * 16 + r) * 128 + kk + 1];
      a1.x = swp[((bt0 + 1) * 16 + r) * 128 + kk];
      a1.y = swp[((bt0 + 1) * 16 + r) * 128 + kk + 1];
      acc0 = wmma_f32(a0, bf, acc0);
      acc1 = wmma_f32(a1, bf, acc1);
    }
    __syncthreads();
  }
  const float invg = 1.f / (gsum[0] + 1e-5f);
  const int half = lane >> 4;
#pragma unroll
  for (int j = 0; j < 8; ++j) {
    const int brow0 = bt0 * 16 + j + half * 8;
    const int brow1 = (bt0 + 1) * 16 + j + half * 8;
    atomicAdd(&out[brow0 * (Rn * Mn) + rhead * Mn + c0 + r], acc0[j] * invg);
    atomicAdd(&out[brow1 * (Rn * Mn) + rhead * Mn + c0 + r], acc1[j] * invg);
  }
}

__global__ void zero_kernel(float* __restrict__ p, int n) {
  const int i = blockIdx.x * 256 + threadIdx.x;
  if (i < n) p[i] = 0.f;
}

// ---------------------------------------------------------------------------
extern "C" void kernel_launch(void* const* d_in, const int* in_sizes, int n_in,
                              void* d_out, int out_size, void* d_ws,
                              size_t ws_size, hipStream_t stream) {
  const float* h_t    = (const float*)d_in[0];
  const float* ww     = (const float*)d_in[1];
  const float* wr     = (const float*)d_in[2];
  const float* memory = (const float*)d_in[3];
  const float* key_w  = (const float*)d_in[4];
  const float* key_b  = (const float*)d_in[5];
  const float* beta_w = (const float*)d_in[6];
  const float* beta_b = (const float*)d_in[7];
  const float* gate_w = (const float*)d_in[8];
  const float* gate_b = (const float*)d_in[9];
  const float* shift_w = (const float*)d_in[10];
  const float* shift_b = (const float*)d_in[11];
  const float* gamma_w = (const float*)d_in[12];
  const float* gamma_b = (const float*)d_in[13];
  const float* erase_w = (const float*)d_in[14];
  const float* erase_b = (const float*)d_in[15];
  const float* add_w   = (const float*)d_in[16];
  const float* add_b   = (const float*)d_in[17];
  const int*   bank_no = (const int*)d_in[18];
  float* out = (float*)d_out;
  float* ws = (float*)d_ws;

  float* kn      = ws + 0;
  float* beta    = ws + 4096;
  float* g       = ws + 4160;
  float* gamma   = ws + 4224;
  float* s       = ws + 4288;
  float* e       = ws + 4480;
  float* a       = ws + 8576;
  float* rowmax  = ws + 12672;
  float* rowsum  = ws + 12736;
  float* gsum    = ws + 12800;
  float* part    = ws + 12832;   // [64*16]
  float* gpart   = ws + 13856;   // [1024]
  float* invnorm = ws + 16384;   // [N]
  float* zbuf    = ws + 81920;   // [B*N] (p stored in place)
  float* wpow    = ws + 4276224; // [B*N]
  float* m1      = ws + 8470528; // [N*M]

  const dim3 rgrid(16, 64);

  head_params_kernel<<<64, 256, 0, stream>>>(
      h_t, key_w, key_b, beta_w, beta_b, gate_w, gate_b, shift_w, shift_b,
      gamma_w, gamma_b, erase_w, erase_b, add_w, add_b, kn, beta, g, gamma, s,
      e, a);

  // ---- write head: weight update on m0 ----
  rownorm_kernel<<<1024, 256, 0, stream>>>(memory, bank_no, invnorm);
  z_wmma_kernel<<<512, 256, 0, stream>>>(memory, bank_no, kn, invnorm, beta,
                                         zbuf);
  rowmax_partial_kernel<<<rgrid, 256, 0, stream>>>(zbuf, part);
  row_finalize_kernel<<<1, 64, 0, stream>>>(part, rowmax, 0);
  expsum_kernel<<<rgrid, 256, 0, stream>>>(zbuf, rowmax, part);
  row_finalize_kernel<<<1, 64, 0, stream>>>(part, rowsum, 1);
  shift_pow_kernel<<<rgrid, 256, 0, stream>>>(zbuf, ww, g, s, gamma, rowsum,
                                              wpow, gpart);
  gsum_kernel<<<1, 256, 0, stream>>>(gpart, gsum);

  // ---- memory erase/add ----
  mem_update_kernel<<<1024, 256, 0, stream>>>(memory, bank_no, wpow, gsum, e, a,
                                              m1);

  // ---- read heads: z/softmax on m1 computed ONCE (shared across R heads) ----
  zero_kernel<<<64, 256, 0, stream>>>(out, out_size);
  rownorm_kernel<<<1024, 256, 0, stream>>>(m1, nullptr, invnorm);
  z_wmma_kernel<<<512, 256, 0, stream>>>(m1, nullptr, kn, invnorm, beta, zbuf);
  rowmax_partial_kernel<<<rgrid, 256, 0, stream>>>(zbuf, part);
  row_finalize_kernel<<<1, 64, 0, stream>>>(part, rowmax, 0);
  expsum_kernel<<<rgrid, 256, 0, stream>>>(zbuf, rowmax, part);
  row_finalize_kernel<<<1, 64, 0, stream>>>(part, rowsum, 1);

  for (int r = 0; r < Rn; ++r) {
    shift_pow_kernel<<<rgrid, 256, 0, stream>>>(
        zbuf, wr + (size_t)r * Bn * Nn, g, s, gamma, rowsum, wpow, gpart);
    gsum_kernel<<<1, 256, 0, stream>>>(gpart, gsum);
    read_wmma_kernel<<<64, 256, 0, stream>>>(wpow, gsum, m1, out, r);
  }
}